// Unpool_LS_23725399343218
// MI455X (gfx1250) — compile-verified
//
#include <hip/hip_runtime.h>

// Unpool_LS for MI455X (gfx1250).
// Memory-bound streaming op: 320 MiB total traffic -> ~14 us at 23.3 TB/s.
// No matmul content => WMMA not applicable; optimize the data-movement path:
// b128 coalesced loads/stores with non-temporal (TH=NT) cache policy, wave32.

typedef float f4v __attribute__((ext_vector_type(4)));

namespace {
constexpr int Bn = 8, Hh = 256, Ww = 256, Cc = 64;
constexpr int Hp = 128, Wp = 128;
constexpr int FULL_ELEMS = Bn * Hh * Ww * Cc;   // 33,554,432
constexpr int POOL_ELEMS = Bn * Hp * Wp * Cc;   // 8,388,608
constexpr int NTHREADS   = POOL_ELEMS / 4;      // one float4 (4 channels) per thread
}

// Per-channel block computation, exactly matching the JAX reference:
//  - stable descending ranks (jnp.argsort(-x) tie-break: lower index first)
//  - cumavg_k = (cumsum_k(sorted_desc) + p) * (1/(k+2))  [same float assoc order]
//  - repl = max(cumavg); amax = FIRST argmax; replace elements with rank <= amax
__device__ __forceinline__ void unpool_lane(float v0, float v1, float v2, float v3,
                                            float p,
                                            float& o0, float& o1, float& o2, float& o3,
                                            float& repl_out)
{
    // stable descending rank[i] = #{ j : v[j] > v[i]  ||  (v[j]==v[i] && j<i) }
    const int r0 = (v1 >  v0) + (v2 >  v0) + (v3 >  v0);
    const int r1 = (v0 >= v1) + (v2 >  v1) + (v3 >  v1);
    const int r2 = (v0 >= v2) + (v1 >= v2) + (v3 >  v2);
    const int r3 = (v0 >= v3) + (v1 >= v3) + (v2 >= v3);

    // descending sorted values: 5-comparator sorting network (value order only;
    // stability is irrelevant for the values themselves)
    float a = v0, b = v1, c = v2, d = v3;
    float mx, mn;
    mx = fmaxf(a, b); mn = fminf(a, b); a = mx; b = mn;
    mx = fmaxf(c, d); mn = fminf(c, d); c = mx; d = mn;
    mx = fmaxf(a, c); mn = fminf(a, c); a = mx; c = mn;
    mx = fmaxf(b, d); mn = fminf(b, d); b = mx; d = mn;
    mx = fmaxf(b, c); mn = fminf(b, c); b = mx; c = mn;

    // cumsum in the reference's association order, then +p, then * inv_count
    const float t0 = a;
    const float t1 = t0 + b;
    const float t2 = t1 + c;
    const float t3 = t2 + d;
    const float c0 = (t0 + p) * 0.5f;
    const float c1 = (t1 + p) * (1.0f / 3.0f);
    const float c2 = (t2 + p) * 0.25f;
    const float c3 = (t3 + p) * 0.2f;

    // first-occurrence argmax (strict '>' keeps the earliest max)
    float m = c0; int am = 0;
    if (c1 > m) { m = c1; am = 1; }
    if (c2 > m) { m = c2; am = 2; }
    if (c3 > m) { m = c3; am = 3; }

    repl_out = m;
    o0 = (r0 <= am) ? m : v0;
    o1 = (r1 <= am) ? m : v1;
    o2 = (r2 <= am) ? m : v2;
    o3 = (r3 <= am) ? m : v3;
}

__global__ __launch_bounds__(256) void unpool_ls_kernel(
    const float* __restrict__ xf,   // x_full   (8,256,256,64)
    const float* __restrict__ xp,   // x_pooled (8,128,128,64)
    float* __restrict__ of,         // output   (8,256,256,64)
    float* __restrict__ orp)        // repl     (8,128,128,64)
{
    const int tid = blockIdx.x * blockDim.x + threadIdx.x;
    if (tid >= NTHREADS) return;

    // tid -> (b, hp, wp, channel-group); channel innermost => coalesced b128
    const int cg = tid & 15;                 // 16 groups of 4 channels
    const int wp = (tid >> 4)  & (Wp - 1);
    const int hp = (tid >> 11) & (Hp - 1);
    const int b  =  tid >> 18;

    const int full_base = (((b * Hh + 2 * hp) * Ww + 2 * wp) * Cc) + cg * 4;
    const int pool_base = tid * 4;           // pooled layout flattens to exactly tid*4

    // streaming reads, zero reuse -> non-temporal policy (gfx1250 TH=NT)
    const f4v v00 = __builtin_nontemporal_load((const f4v*)(xf + full_base));                 // (dh0,dw0)
    const f4v v01 = __builtin_nontemporal_load((const f4v*)(xf + full_base + Cc));            // (dh0,dw1)
    const f4v v10 = __builtin_nontemporal_load((const f4v*)(xf + full_base + Ww * Cc));       // (dh1,dw0)
    const f4v v11 = __builtin_nontemporal_load((const f4v*)(xf + full_base + Ww * Cc + Cc));  // (dh1,dw1)
    const f4v pv  = __builtin_nontemporal_load((const f4v*)(xp + pool_base));

    f4v o00, o01, o10, o11, rp;
#pragma unroll
    for (int l = 0; l < 4; ++l) {
        float a0, a1, a2, a3, rr;
        unpool_lane(v00[l], v01[l], v10[l], v11[l], pv[l], a0, a1, a2, a3, rr);
        o00[l] = a0; o01[l] = a1; o10[l] = a2; o11[l] = a3; rp[l] = rr;
    }

    __builtin_nontemporal_store(o00, (f4v*)(of + full_base));
    __builtin_nontemporal_store(o01, (f4v*)(of + full_base + Cc));
    __builtin_nontemporal_store(o10, (f4v*)(of + full_base + Ww * Cc));
    __builtin_nontemporal_store(o11, (f4v*)(of + full_base + Ww * Cc + Cc));
    __builtin_nontemporal_store(rp,  (f4v*)(orp + pool_base));
}

extern "C" void kernel_launch(void* const* d_in, const int* in_sizes, int n_in,
                              void* d_out, int out_size, void* d_ws, size_t ws_size,
                              hipStream_t stream) {
    const float* xf = (const float*)d_in[0];   // x_full
    const float* xp = (const float*)d_in[1];   // x_pooled
    float* of  = (float*)d_out;                // output (first tuple element)
    float* orp = (float*)d_out + FULL_ELEMS;   // repl   (second tuple element)

    const int threads = 256;                   // 8 wave32 waves per workgroup
    const int blocks  = (NTHREADS + threads - 1) / threads;  // 8192
    unpool_ls_kernel<<<blocks, threads, 0, stream>>>(xf, xp, of, orp);
}